// Module_9328668967505
// MI455X (gfx1250) — compile-verified
//
#include <hip/hip_runtime.h>
#include <math.h>

// Per-face Gaussian parameterization from a triangle soup.
// Output row (14 floats): mean(3) | quat WXYZ(4) | scale(3) | color(3) | opacity(1)

namespace {

constexpr float kEps = 1e-12f;

// Native clang vector type: accepted by __builtin_nontemporal_store.
typedef __attribute__((ext_vector_type(2))) float v2f;

// Hardware-rate helpers (v_rcp_f32 / v_rsq_f32 / v_sqrt_f32, ~1 ulp).
__device__ __forceinline__ float fast_rcp(float x)  { return __builtin_amdgcn_rcpf(x); }
__device__ __forceinline__ float fast_rsq(float x)  { return __builtin_amdgcn_rsqf(x); }
__device__ __forceinline__ float fast_sqrt(float x) { return __builtin_amdgcn_sqrtf(x); }

// smooth_clamp with slope_l == 0 (all three call sites use slope_l = 0).
// inv_w = 1/(rb-lb) precomputed at compile time.
__device__ __forceinline__ float smooth_clamp_r(float x, float lb, float rb,
                                                float inv_w, float sr) {
  float t    = (x - lb) * inv_w;
  float e    = __expf(2.0f - 4.0f * t);          // v_exp_f32 path
  float sig  = fast_rcp(1.0f + e);               // sigmoid(4t - 2)
  float core = lb + (rb - lb) * sig;
  return core + sr * fmaxf(x - rb, 0.0f);
}

__global__ __launch_bounds__(256) void face_gaussian_kernel(
    const float* __restrict__ va, const float* __restrict__ vb,
    const float* __restrict__ vc, const float* __restrict__ rotz,
    const float* __restrict__ gsx, const float* __restrict__ gsy,
    const float* __restrict__ gsz, const float* __restrict__ col,
    const float* __restrict__ lop, float* __restrict__ out, int F) {
  int i = blockIdx.x * blockDim.x + threadIdx.x;
  if (i >= F) return;

  const long b3 = 3L * (long)i;

  // Streamed once: non-temporal loads (TH=NT) — don't rinse L2.
  float aX = __builtin_nontemporal_load(va + b3 + 0);
  float aY = __builtin_nontemporal_load(va + b3 + 1);
  float aZ = __builtin_nontemporal_load(va + b3 + 2);
  float bX = __builtin_nontemporal_load(vb + b3 + 0);
  float bY = __builtin_nontemporal_load(vb + b3 + 1);
  float bZ = __builtin_nontemporal_load(vb + b3 + 2);
  float cX = __builtin_nontemporal_load(vc + b3 + 0);
  float cY = __builtin_nontemporal_load(vc + b3 + 1);
  float cZ = __builtin_nontemporal_load(vc + b3 + 2);

  float rz  = __builtin_nontemporal_load(rotz + i);
  float sxl = __builtin_nontemporal_load(gsx + i);
  float syl = __builtin_nontemporal_load(gsy + i);
  float szl = __builtin_nontemporal_load(gsz + i);
  float cr  = __builtin_nontemporal_load(col + b3 + 0);
  float cg  = __builtin_nontemporal_load(col + b3 + 1);
  float cb  = __builtin_nontemporal_load(col + b3 + 2);
  float lo  = __builtin_nontemporal_load(lop + i);

  // --- face frame ---
  float e1x = bX - aX, e1y = bY - aY, e1z = bZ - aZ;
  float e2x = cX - aX, e2y = cY - aY, e2z = cZ - aZ;

  float nx = e1y * e2z - e1z * e2y;
  float ny = e1z * e2x - e1x * e2z;
  float nz = e1x * e2y - e1y * e2x;

  float nn   = fast_sqrt(nx * nx + ny * ny + nz * nz);
  float area = 0.5f * nn;
  float e1n  = fast_sqrt(e1x * e1x + e1y * e1y + e1z * e1z);

  float inv1 = fast_rcp(e1n + kEps);
  float axx = e1x * inv1, axy = e1y * inv1, axz = e1z * inv1;
  float invn = fast_rcp(nn + kEps);
  float azx = nx * invn, azy = ny * invn, azz = nz * invn;

  // ay = az x ax
  float ayx = azy * axz - azz * axy;
  float ayy = azz * axx - azx * axz;
  float ayz = azx * axy - azy * axx;

  // centroid
  constexpr float kThird = 1.0f / 3.0f;
  float tx = (aX + bX + cX) * kThird;
  float ty = (aY + bY + cY) * kThird;
  float tz = (aZ + bZ + cZ) * kThird;

  // R columns = (ax, ay, az)
  float m00 = axx, m01 = ayx, m02 = azx;
  float m10 = axy, m11 = ayy, m12 = azy;
  float m20 = axz, m21 = ayz, m22 = azz;
  float tr = m00 + m11 + m22;

  // robust 4-case rot-mat -> quat (WXYZ), argmax-first tie semantics
  float qw, qx, qy, qz;
  if (tr >= m00 && tr >= m11 && tr >= m22) {
    float t0 = fmaxf(1.0f + tr, 1e-8f);
    float s = 0.5f * fast_rsq(t0);
    qw = t0 * s; qx = (m21 - m12) * s; qy = (m02 - m20) * s; qz = (m10 - m01) * s;
  } else if (m00 >= m11 && m00 >= m22) {
    float t1 = fmaxf(1.0f + m00 - m11 - m22, 1e-8f);
    float s = 0.5f * fast_rsq(t1);
    qw = (m21 - m12) * s; qx = t1 * s; qy = (m01 + m10) * s; qz = (m02 + m20) * s;
  } else if (m11 >= m22) {
    float t2 = fmaxf(1.0f - m00 + m11 - m22, 1e-8f);
    float s = 0.5f * fast_rsq(t2);
    qw = (m02 - m20) * s; qx = (m01 + m10) * s; qy = t2 * s; qz = (m12 + m21) * s;
  } else {
    float t3 = fmaxf(1.0f - m00 - m11 + m22, 1e-8f);
    float s = 0.5f * fast_rsq(t3);
    qw = (m10 - m01) * s; qx = (m02 + m20) * s; qy = (m12 + m21) * s; qz = t3 * s;
  }

  // world_q = face_q * local_q, local_q = (cos(h), 0, 0, sin(h))
  float half = rz * 0.5f;
  float sh, ch;
  __sincosf(half, &sh, &ch);   // native v_sin_f32 / v_cos_f32; |half| is small
  float wq = qw * ch - qz * sh;
  float wx = qx * ch + qy * sh;
  float wy = qy * ch - qx * sh;
  float wz = qw * sh + qz * ch;

  // --- scales ---
  float sf = fast_sqrt(area);
  float sx = sf * smooth_clamp_r(__expf(sxl), 0.1f, 3.0f, 1.0f / 2.9f, 0.005f);
  float sy = sf * smooth_clamp_r(__expf(syl), 0.1f, 3.0f, 1.0f / 2.9f, 0.005f);
  float sz = smooth_clamp_r(sf * __expf(szl), 0.001f, 0.01f, 1.0f / 0.009f, 0.005f);

  float op = __expf(lo);

  // Output row: 56-byte pitch -> 8B-aligned; 7 non-temporal b64 stores.
  v2f* o2 = reinterpret_cast<v2f*>(out + 14L * (long)i);
  __builtin_nontemporal_store((v2f){tx, ty}, o2 + 0);
  __builtin_nontemporal_store((v2f){tz, wq}, o2 + 1);
  __builtin_nontemporal_store((v2f){wx, wy}, o2 + 2);
  __builtin_nontemporal_store((v2f){wz, sx}, o2 + 3);
  __builtin_nontemporal_store((v2f){sy, sz}, o2 + 4);
  __builtin_nontemporal_store((v2f){cr, cg}, o2 + 5);
  __builtin_nontemporal_store((v2f){cb, op}, o2 + 6);
}

}  // namespace

extern "C" void kernel_launch(void* const* d_in, const int* in_sizes, int n_in,
                              void* d_out, int out_size, void* d_ws, size_t ws_size,
                              hipStream_t stream) {
  const float* va  = (const float*)d_in[0];
  const float* vb  = (const float*)d_in[1];
  const float* vc  = (const float*)d_in[2];
  const float* rz  = (const float*)d_in[3];
  const float* gsx = (const float*)d_in[4];
  const float* gsy = (const float*)d_in[5];
  const float* gsz = (const float*)d_in[6];
  const float* col = (const float*)d_in[7];
  const float* lop = (const float*)d_in[8];

  const int F = in_sizes[3];  // gp_rot_z is [F]
  const int threads = 256;    // 8 wave32 per block
  const int blocks = (F + threads - 1) / threads;

  face_gaussian_kernel<<<blocks, threads, 0, stream>>>(
      va, vb, vc, rz, gsx, gsy, gsz, col, lop, (float*)d_out, F);
}